// KNNCRFLayer_12300786336176
// MI455X (gfx1250) — compile-verified
//
#include <hip/hip_runtime.h>
#include <hip/hip_bf16.h>
#include <math.h>

#define KNBR 16
#define NCLS 20

typedef __attribute__((ext_vector_type(2))) float v2f;
typedef __attribute__((ext_vector_type(4))) float v4f;
typedef __attribute__((ext_vector_type(8))) float v8f;

#define BIGF 3.4e38f

// ---------------------------------------------------------------------------
// Kernel 1: squared norms of coords
// ---------------------------------------------------------------------------
__global__ void sqnorm_kernel(const float* __restrict__ coords,
                              float* __restrict__ sq, int n) {
    int i = blockIdx.x * blockDim.x + threadIdx.x;
    if (i < n) {
        float x = coords[3 * i + 0];
        float y = coords[3 * i + 1];
        float z = coords[3 * i + 2];
        sq[i] = x * x + y * y + z * z;
    }
}

// ---------------------------------------------------------------------------
// Kernel 2: KNN. One wave32 handles 32 query rows (two 16-row WMMA A-tiles).
// Per candidate tile: one B load -> two V_WMMA_F32_16X16X4_F32 -> d2 tiles in
// LDS -> every lane owns one query row and updates an in-register top-16.
// ---------------------------------------------------------------------------
__global__ __launch_bounds__(32) void knn_kernel(const float* __restrict__ coords,
                                                 const float* __restrict__ sq,
                                                 int* __restrict__ knn_out,
                                                 int n) {
    // row stride 20 floats: keeps each row's 16 candidates 16B-aligned
    __shared__ float d2s[32][20];
    __shared__ float sqq_s[32];

    const int lane  = threadIdx.x;          // 0..31
    const int m     = lane & 15;
    const int hi    = lane >> 4;            // 0: lanes 0-15, 1: lanes 16-31
    const int qbase = blockIdx.x * 32;

    // ---- A matrices (two 16x4 fp32 query tiles).
    // Lane layout: row M = lane%16; lanes 0-15 hold (K0,K1)=(x,y),
    // lanes 16-31 hold (K2,K3)=(z,0).  Branchless via selects.
    int q0 = qbase + m;       if (q0 >= n) q0 = n - 1;
    int q1 = qbase + 16 + m;  if (q1 >= n) q1 = n - 1;
    float q0x = coords[q0 * 3 + 0], q0y = coords[q0 * 3 + 1], q0z = coords[q0 * 3 + 2];
    float q1x = coords[q1 * 3 + 0], q1y = coords[q1 * 3 + 1], q1z = coords[q1 * 3 + 2];
    v2f a0, a1;
    a0.x = hi ? q0z : q0x;  a0.y = hi ? 0.0f : q0y;
    a1.x = hi ? q1z : q1x;  a1.y = hi ? 0.0f : q1y;

    {
        int ql = qbase + lane; if (ql >= n) ql = n - 1;
        sqq_s[lane] = sq[ql];
    }
    __syncthreads();

    // ---- per-row top-16 state (lane owns query row = lane)
    float bd[KNBR];
    int   bi[KNBR];
#pragma unroll
    for (int s = 0; s < KNBR; ++s) { bd[s] = BIGF; bi[s] = 0; }
    float worst = BIGF;

    for (int cb = 0; cb < n; cb += 16) {
        // prefetch next candidate tile (speculative, WGP scope)
        {
            int pj = cb + 16; if (pj >= n) pj = n - 1;
            __builtin_prefetch(coords + (size_t)pj * 3, 0, 0);
        }

        // ---- B matrix (4x16 fp32 candidate tile): lane holds column N=lane%16
        int cj = cb + m;
        const bool colvalid = (cb + m) < n;
        if (cj >= n) cj = n - 1;
        float cx = coords[cj * 3 + 0], cy = coords[cj * 3 + 1], cz = coords[cj * 3 + 2];
        v2f b;
        b.x = hi ? cz : cx;  b.y = hi ? 0.0f : cy;
        float sqc = sq[cj];

        v8f c0 = {}, c1 = {};
        c0 = __builtin_amdgcn_wmma_f32_16x16x4_f32(false, a0, false, b,
                                                   (short)0, c0, false, false);
        c1 = __builtin_amdgcn_wmma_f32_16x16x4_f32(false, a1, false, b,
                                                   (short)0, c1, false, false);

        __syncthreads();   // previous tile's selection reads complete
        // C layout: VGPR r -> row (r + 8*hi), column N = lane%16
#pragma unroll
        for (int r = 0; r < 8; ++r) {
            int row0 = r + 8 * hi;
            float d0 = sqq_s[row0]      + sqc - 2.0f * c0[r];
            float d1 = sqq_s[row0 + 16] + sqc - 2.0f * c1[r];
            d2s[row0][m]      = colvalid ? d0 : BIGF;
            d2s[row0 + 16][m] = colvalid ? d1 : BIGF;
        }
        __syncthreads();

        // ---- selection: every lane scans its row's 16 candidates.
        float dreg[16];
#pragma unroll
        for (int jj = 0; jj < 4; ++jj) {
            v4f dv = *(const v4f*)&d2s[lane][jj * 4];
            dreg[jj * 4 + 0] = dv.x;
            dreg[jj * 4 + 1] = dv.y;
            dreg[jj * 4 + 2] = dv.z;
            dreg[jj * 4 + 3] = dv.w;
        }
#pragma unroll
        for (int j = 0; j < 16; ++j) {
            float d    = dreg[j];
            int   cidx = cb + j;
            if (d < worst) {
                int   ws = 0;
                float wv = bd[0];
#pragma unroll
                for (int s = 1; s < KNBR; ++s)
                    if (bd[s] > wv) { wv = bd[s]; ws = s; }
#pragma unroll
                for (int s = 0; s < KNBR; ++s)
                    if (s == ws) { bd[s] = d; bi[s] = cidx; }
                wv = bd[0];
#pragma unroll
                for (int s = 1; s < KNBR; ++s) wv = fmaxf(wv, bd[s]);
                worst = wv;
            }
        }
    }

    if (qbase + lane < n) {
        int* out = knn_out + (size_t)(qbase + lane) * KNBR;
#pragma unroll
        for (int s = 0; s < KNBR; ++s) out[s] = bi[s];
    }
}

// ---------------------------------------------------------------------------
// Kernel 3: initial softmax of logits -> q0
// ---------------------------------------------------------------------------
__global__ void softmax_kernel(const float* __restrict__ logits,
                               float* __restrict__ q, int n) {
    int i = blockIdx.x * blockDim.x + threadIdx.x;
    if (i >= n) return;
    float v[NCLS];
    float mx = -BIGF;
#pragma unroll
    for (int c = 0; c < NCLS; ++c) { v[c] = logits[i * NCLS + c]; mx = fmaxf(mx, v[c]); }
    float s = 0.0f;
#pragma unroll
    for (int c = 0; c < NCLS; ++c) { v[c] = __expf(v[c] - mx); s += v[c]; }
    float inv = 1.0f / s;
#pragma unroll
    for (int c = 0; c < NCLS; ++c) q[i * NCLS + c] = v[c] * inv;
}

// ---------------------------------------------------------------------------
// Kernel 4: one CRF mean-field iteration.
// msg = mean_k q[knn[i,k]]; refined = logits + msg @ W^T; qout = softmax(refined)
// ---------------------------------------------------------------------------
__global__ void crf_iter_kernel(const float* __restrict__ logits,
                                const float* __restrict__ qin,
                                const int* __restrict__ knn,
                                const float* __restrict__ W,   // 20x20 row-major
                                float* __restrict__ qout,
                                float* __restrict__ refined_out,
                                int n) {
    int i = blockIdx.x * blockDim.x + threadIdx.x;
    if (i >= n) return;

    float msg[NCLS];
#pragma unroll
    for (int c = 0; c < NCLS; ++c) msg[c] = 0.0f;

    const int* nb = knn + (size_t)i * KNBR;
#pragma unroll 4
    for (int k = 0; k < KNBR; ++k) {
        const float* qj = qin + (size_t)nb[k] * NCLS;
#pragma unroll
        for (int c = 0; c < NCLS; ++c) msg[c] += qj[c];
    }
    const float invk = 1.0f / (float)KNBR;
#pragma unroll
    for (int c = 0; c < NCLS; ++c) msg[c] *= invk;

    float r[NCLS];
    float mx = -BIGF;
#pragma unroll
    for (int c = 0; c < NCLS; ++c) {
        float acc = logits[i * NCLS + c];
        const float* wr = W + c * NCLS;   // refined[c] = logits[c] + sum_d msg[d]*W[c][d]
#pragma unroll
        for (int d = 0; d < NCLS; ++d) acc = fmaf(msg[d], wr[d], acc);
        r[c] = acc;
        mx = fmaxf(mx, acc);
    }
#pragma unroll
    for (int c = 0; c < NCLS; ++c) refined_out[i * NCLS + c] = r[c];

    float s = 0.0f;
    float e[NCLS];
#pragma unroll
    for (int c = 0; c < NCLS; ++c) { e[c] = __expf(r[c] - mx); s += e[c]; }
    float inv = 1.0f / s;
#pragma unroll
    for (int c = 0; c < NCLS; ++c) qout[i * NCLS + c] = e[c] * inv;
}

// ---------------------------------------------------------------------------
// Launcher
// ---------------------------------------------------------------------------
static inline size_t align256(size_t x) { return (x + 255) & ~(size_t)255; }

extern "C" void kernel_launch(void* const* d_in, const int* in_sizes, int n_in,
                              void* d_out, int out_size, void* d_ws, size_t ws_size,
                              hipStream_t stream) {
    const float* logits = (const float*)d_in[0];   // (N, 20)
    const float* coords = (const float*)d_in[1];   // (N, 3)
    const float* W      = (const float*)d_in[2];   // (20, 20)
    const int n = in_sizes[1] / 3;

    // workspace layout
    char*  ws   = (char*)d_ws;
    size_t off  = 0;
    float* sq   = (float*)(ws + off);  off += align256((size_t)n * sizeof(float));
    int*   knn  = (int*)  (ws + off);  off += align256((size_t)n * KNBR * sizeof(int));
    float* qA   = (float*)(ws + off);  off += align256((size_t)n * NCLS * sizeof(float));
    float* qB   = (float*)(ws + off);  off += align256((size_t)n * NCLS * sizeof(float));

    float* refined = (float*)d_out;                      // first output
    float* q_final = (float*)d_out + (size_t)n * NCLS;   // second output

    const int T = 256;
    const int nb = (n + T - 1) / T;

    sqnorm_kernel<<<nb, T, 0, stream>>>(coords, sq, n);

    const int qtiles = (n + 31) / 32;
    knn_kernel<<<qtiles, 32, 0, stream>>>(coords, sq, knn, n);

    softmax_kernel<<<nb, T, 0, stream>>>(logits, qA, n);

    // 3 mean-field iterations; refined is overwritten each time (last wins)
    crf_iter_kernel<<<nb, T, 0, stream>>>(logits, qA, knn, W, qB, refined, n);
    crf_iter_kernel<<<nb, T, 0, stream>>>(logits, qB, knn, W, qA, refined, n);
    crf_iter_kernel<<<nb, T, 0, stream>>>(logits, qA, knn, W, q_final, refined, n);
}